// GraphNorm_62749472195049
// MI455X (gfx1250) — compile-verified
//
#include <hip/hip_runtime.h>

// GraphNorm for MI455X (gfx1250).
// Memory-bound (512 MB min traffic -> ~22us @ 23.3 TB/s). Strategy:
//   - 1 workgroup per graph; whole 256KB graph tile DMA'd into LDS via the
//     Tensor Data Mover (tensor_load_to_lds, TENSORcnt), 320KB LDS/WGP fits it.
//   - single LDS sweep accumulates per-feature sum and sum-of-squares;
//     var = E[x^2] - 2*m*E[x] + m^2 with m = mean_scale * mean  (matches the
//     reference, which scales the mean before centering AND variance).
//   - normalize sweep reads LDS, writes coalesced non-temporal b128 stores.
// Net HBM traffic: x read once, y written once.

#define NPG    512          // nodes per graph
#define DFEAT  128          // feature dim
#define EPSV   1e-6f
#define BLOCK  512          // 16 wave32s
#define NQ     (DFEAT / 4)  // 32 feature-quads (float4)
#define NGRP   (BLOCK / NQ) // 16 row-groups
#define RPG    (NPG / NGRP) // 32 rows per group

typedef __attribute__((ext_vector_type(4))) float        f32x4;
typedef __attribute__((ext_vector_type(4))) unsigned int u32x4;
typedef __attribute__((ext_vector_type(8))) int          i32x8;
typedef __attribute__((ext_vector_type(4))) int          i32x4;

__global__ __launch_bounds__(BLOCK) void graphnorm_tdm_kernel(
    const float* __restrict__ x, const int* __restrict__ n_node,
    const float* __restrict__ mean_scale, const float* __restrict__ scale,
    const float* __restrict__ bias, float* __restrict__ out)
{
    __shared__ f32x4 tile4[NPG * NQ];      // 256 KB: the whole graph's x
    __shared__ f32x4 red_s[NGRP][NQ];      // 8 KB partial sums
    __shared__ f32x4 red_ss[NGRP][NQ];     // 8 KB partial sum-of-squares
    __shared__ f32x4 bc_m[NQ];             // broadcast: scaled mean
    __shared__ f32x4 bc_mul[NQ];           // broadcast: rsqrt(var+eps)*scale

    const int g   = blockIdx.x;
    const int tid = threadIdx.x;
    const int q   = tid & (NQ - 1);     // feature quad 0..31
    const int p   = tid >> 5;           // row group 0..15

    const size_t gbase = (size_t)g * NPG * DFEAT;   // in floats

    // ---------------- TDM: Global -> LDS DMA of the 512x128 f32 tile --------
    if (tid < 32) {  // one wave issues the descriptor (TDM ignores EXEC)
        unsigned long long ga =
            (unsigned long long)(const void*)(x + gbase);
        unsigned int lds0 =
            (unsigned int)(unsigned long long)(void*)&tile4[0]; // low 32b of
                                                                // generic ptr
                                                                // = LDS offset
        // D# group 0 (bits per CDNA5 ISA)
        u32x4 d0;
        d0[0] = 1u;                                  // count=1, user mode
        d0[1] = lds0;                                // lds_addr
        d0[2] = (unsigned int)ga;                    // global_addr[31:0]
        d0[3] = ((unsigned int)(ga >> 32) & 0x01FFFFFFu)  // global_addr[56:32]
                | (2u << 30);                        // type = 2 ("image")
        // D# group 1 (§8.4): 2-D tensor, data_size=4B, tile 128x512
        i32x8 d1;
        d1[0] = (2 << 16);                 // workgroup_mask=0, data_size=4B
        d1[1] = (DFEAT & 0xFFFF) << 16;    // tensor_dim0[15:0] = 128
        d1[2] = (NPG & 0xFFFF) << 16;      // tensor_dim0 hi=0 | tensor_dim1=512
        d1[3] = (DFEAT << 16);             // tensor_dim1 hi=0 | tile_dim0=128
        d1[4] = NPG;                       // tile_dim1=512, tile_dim2=0
        d1[5] = DFEAT;                     // tensor_dim0_stride = 128 elems
        d1[6] = 0;                         // stride hi | tensor_dim1_stride lo
        d1[7] = 0;
        i32x4 dz4 = {0, 0, 0, 0};          // groups 2/3 unused (<=2D)
        i32x8 dz8 = {0, 0, 0, 0, 0, 0, 0, 0};
        __builtin_amdgcn_tensor_load_to_lds(d0, d1, dz4, dz4, dz8, 0);
        __builtin_amdgcn_s_wait_tensorcnt(0);
    }
    __syncthreads();   // LDS tile now visible to all 16 waves

    // ---------------- one LDS sweep: per-feature sum and sum of squares -----
    float sx = 0.f, sy = 0.f, sz = 0.f, sw = 0.f;
    float qx = 0.f, qy = 0.f, qz = 0.f, qw = 0.f;
    const f32x4* trow = &tile4[(size_t)p * RPG * NQ + q];
    #pragma unroll 8
    for (int n = 0; n < RPG; ++n) {
        f32x4 v = trow[n * NQ];
        sx += v.x; sy += v.y; sz += v.z; sw += v.w;
        qx = fmaf(v.x, v.x, qx); qy = fmaf(v.y, v.y, qy);
        qz = fmaf(v.z, v.z, qz); qw = fmaf(v.w, v.w, qw);
    }
    {
        f32x4 s4; s4.x = sx; s4.y = sy; s4.z = sz; s4.w = sw;
        f32x4 q4; q4.x = qx; q4.y = qy; q4.z = qz; q4.w = qw;
        red_s[p][q]  = s4;
        red_ss[p][q] = q4;
    }
    __syncthreads();

    // ---------------- combine 16 partials, derive m and mul per feature -----
    if (p == 0) {
        float Sx = 0.f, Sy = 0.f, Sz = 0.f, Sw = 0.f;
        float Qx = 0.f, Qy = 0.f, Qz = 0.f, Qw = 0.f;
        #pragma unroll
        for (int r = 0; r < NGRP; ++r) {
            f32x4 a = red_s[r][q];  Sx += a.x; Sy += a.y; Sz += a.z; Sw += a.w;
            f32x4 b = red_ss[r][q]; Qx += b.x; Qy += b.y; Qz += b.z; Qw += b.w;
        }
        const float inv_n = 1.0f / (float)n_node[g];
        const f32x4 msc = ((const f32x4*)mean_scale)[q];
        const f32x4 scl = ((const f32x4*)scale)[q];
        float mean, m, var;
        f32x4 bm, bmul;
        mean = Sx * inv_n; m = mean * msc.x;
        var  = fmaf(m, m, fmaf(-2.0f * m, mean, Qx * inv_n));
        bm.x = m; bmul.x = rsqrtf(var + EPSV) * scl.x;
        mean = Sy * inv_n; m = mean * msc.y;
        var  = fmaf(m, m, fmaf(-2.0f * m, mean, Qy * inv_n));
        bm.y = m; bmul.y = rsqrtf(var + EPSV) * scl.y;
        mean = Sz * inv_n; m = mean * msc.z;
        var  = fmaf(m, m, fmaf(-2.0f * m, mean, Qz * inv_n));
        bm.z = m; bmul.z = rsqrtf(var + EPSV) * scl.z;
        mean = Sw * inv_n; m = mean * msc.w;
        var  = fmaf(m, m, fmaf(-2.0f * m, mean, Qw * inv_n));
        bm.w = m; bmul.w = rsqrtf(var + EPSV) * scl.w;
        bc_m[q]   = bm;
        bc_mul[q] = bmul;
    }
    __syncthreads();

    // ---------------- normalize sweep: LDS -> coalesced NT b128 stores ------
    const f32x4 m4   = bc_m[q];
    const f32x4 mul4 = bc_mul[q];
    const f32x4 b4   = ((const f32x4*)bias)[q];
    f32x4* orow = (f32x4*)(out + gbase) + (size_t)p * RPG * NQ + q;
    #pragma unroll 8
    for (int n = 0; n < RPG; ++n) {
        f32x4 v = trow[n * NQ];
        f32x4 r;
        r.x = fmaf(v.x - m4.x, mul4.x, b4.x);
        r.y = fmaf(v.y - m4.y, mul4.y, b4.y);
        r.z = fmaf(v.z - m4.z, mul4.z, b4.z);
        r.w = fmaf(v.w - m4.w, mul4.w, b4.w);
        __builtin_nontemporal_store(r, &orow[n * NQ]);
    }
}

extern "C" void kernel_launch(void* const* d_in, const int* in_sizes, int n_in,
                              void* d_out, int out_size, void* d_ws, size_t ws_size,
                              hipStream_t stream) {
    const float* x          = (const float*)d_in[0];
    const int*   n_node     = (const int*)d_in[1];
    const float* mean_scale = (const float*)d_in[2];
    const float* scale      = (const float*)d_in[3];
    const float* bias       = (const float*)d_in[4];
    float*       out        = (float*)d_out;

    const int n_graph = in_sizes[1];   // 1024 per reference
    graphnorm_tdm_kernel<<<dim3(n_graph), dim3(BLOCK), 0, stream>>>(
        x, n_node, mean_scale, scale, bias, out);
}